// GNNPlusAct_11081015623988
// MI455X (gfx1250) — compile-verified
//
#include <hip/hip_runtime.h>
#include <hip/hip_bf16.h>

typedef __attribute__((ext_vector_type(2))) float v2f;
typedef __attribute__((ext_vector_type(8))) float v8f;

#define D 128   // D_IN == D_OUT == 128

// ---------------- degree / norm ----------------
__global__ void k_init_deg(float* __restrict__ deg, int N) {
    int i = blockIdx.x * blockDim.x + threadIdx.x;
    if (i < N) deg[i] = 1.0f;              // self loop contributes 1
}

__global__ void k_count_deg(const int* __restrict__ dst, float* __restrict__ deg, int E) {
    int e = blockIdx.x * blockDim.x + threadIdx.x;
    if (e < E) atomicAdd(&deg[dst[e]], 1.0f);
}

__global__ void k_deg_to_dinv(float* __restrict__ deg, int N) {
    int i = blockIdx.x * blockDim.x + threadIdx.x;
    if (i < N) {
        float d = deg[i];
        deg[i] = (d > 0.0f) ? rsqrtf(d) : 0.0f;   // in place: deg -> dinv
    }
}

// ---------------- h = x @ W via fp32 WMMA ----------------
// Block = 256 threads = 8 waves. Block b owns rows [16b, 16b+16).
// Wave w owns output columns [16w, 16w+16). K-loop of 32 x 16x16x4 fp32 WMMA.
__global__ void __launch_bounds__(256)
k_gemm_wmma(const float* __restrict__ x, const float* __restrict__ W,
            float* __restrict__ h, int N) {
    const int lane = threadIdx.x & 31;
    const int wave = threadIdx.x >> 5;
    const int row0 = blockIdx.x * 16;
    const int col0 = wave * 16;
    if (row0 >= N) return;

    const int  ml = lane & 15;          // M (for A) or N (for B/C) within tile
    const bool hi = lane >= 16;         // lanes 16-31 carry K+2 halves

    v8f c = {};
    const float* __restrict__ xrow = x + (size_t)(row0 + ml) * D;

    for (int k = 0; k < D; k += 4) {
        const int ka = k + (hi ? 2 : 0);
        v2f a, bm;
        // A 16x4 fp32: lane<16 -> {K=k,k+1}, lane>=16 -> {K=k+2,k+3} for row M=ml
        a.x = xrow[ka];
        a.y = xrow[ka + 1];
        // B 4x16 fp32: VGPR0 = row K=ka, VGPR1 = row K=ka+1, col N=col0+ml
        bm.x = W[(size_t)ka * D + col0 + ml];
        bm.y = W[(size_t)(ka + 1) * D + col0 + ml];
        c = __builtin_amdgcn_wmma_f32_16x16x4_f32(
                /*neg_a=*/false, a, /*neg_b=*/false, bm,
                /*c_mod=*/(short)0, c, /*reuse_a=*/false, /*reuse_b=*/false);
    }

    // C/D layout: VGPR i -> M = i (lanes 0-15) or M = 8+i (lanes 16-31), N = ml
#pragma unroll
    for (int i = 0; i < 8; ++i) {
        const int m = hi ? (8 + i) : i;
        h[(size_t)(row0 + m) * D + col0 + ml] = c[i];
    }
}

// ---------------- out = b + h * dinv^2 (self-loop term) ----------------
__global__ void k_init_out(const float* __restrict__ h, const float* __restrict__ dinv,
                           const float* __restrict__ bias, float* __restrict__ out, int total) {
    int t = blockIdx.x * blockDim.x + threadIdx.x;
    if (t < total) {
        int i = t >> 7;          // node
        int j = t & (D - 1);     // feature
        float di = dinv[i];
        out[t] = bias[j] + h[t] * di * di;
    }
}

// ---------------- edge scatter: 1 wave per edge, float4 per lane ----------------
__global__ void k_scatter(const float* __restrict__ h, const float* __restrict__ dinv,
                          const int* __restrict__ src, const int* __restrict__ dst,
                          float* __restrict__ out, int E) {
    int t = blockIdx.x * blockDim.x + threadIdx.x;
    int e = t >> 5;
    int lane = t & 31;
    if (e >= E) return;
    int s = src[e];
    int d = dst[e];
    float norm = dinv[s] * dinv[d];
    const float4 v = ((const float4*)(h + (size_t)s * D))[lane];
    float* o = out + (size_t)d * D + lane * 4;
    atomicAdd(o + 0, v.x * norm);
    atomicAdd(o + 1, v.y * norm);
    atomicAdd(o + 2, v.z * norm);
    atomicAdd(o + 3, v.w * norm);
}

// ---------------- PReLU (in place) ----------------
__global__ void k_prelu(float* __restrict__ out, const float* __restrict__ a, int total) {
    int t = blockIdx.x * blockDim.x + threadIdx.x;
    if (t < total) {
        float slope = a[0];
        float v = out[t];
        out[t] = (v >= 0.0f) ? v : slope * v;
    }
}

extern "C" void kernel_launch(void* const* d_in, const int* in_sizes, int n_in,
                              void* d_out, int out_size, void* d_ws, size_t ws_size,
                              hipStream_t stream) {
    const float* x    = (const float*)d_in[0];   // [N, 128]
    const float* W    = (const float*)d_in[1];   // [128, 128]
    const float* bias = (const float*)d_in[2];   // [128]
    const float* a    = (const float*)d_in[3];   // [1]
    const int*   ei   = (const int*)d_in[4];     // [2, E]

    const int N = in_sizes[0] / D;
    const int E = in_sizes[4] / 2;
    const int* src = ei;        // edge_index[0]
    const int* dst = ei + E;    // edge_index[1]

    float* dinv = (float*)d_ws;      // N floats (deg then dinv, in place)
    float* h    = dinv + N;          // N*128 floats

    const int total = N * D;

    // 1-3: degree + rsqrt norm
    k_init_deg   <<<(N + 255) / 256, 256, 0, stream>>>(dinv, N);
    k_count_deg  <<<(E + 255) / 256, 256, 0, stream>>>(dst, dinv, E);
    k_deg_to_dinv<<<(N + 255) / 256, 256, 0, stream>>>(dinv, N);

    // 4: h = x @ W  (N divisible by 16 -> exact tiling; guard kept anyway)
    k_gemm_wmma<<<(N + 15) / 16, 256, 0, stream>>>(x, W, h, N);

    // 5: out = b + h * dinv^2 (self loop), fully overwrites d_out
    k_init_out<<<(total + 255) / 256, 256, 0, stream>>>(h, dinv, bias, (float*)d_out, total);

    // 6: scatter-add over edges (1 wave / edge)
    {
        long long threads = (long long)E * 32;
        int blocks = (int)((threads + 255) / 256);
        k_scatter<<<blocks, 256, 0, stream>>>(h, dinv, src, dst, (float*)d_out, E);
    }

    // 7: PReLU in place
    k_prelu<<<(total + 255) / 256, 256, 0, stream>>>((float*)d_out, a, total);
}